// AdditiveAttention_49787260895463
// MI455X (gfx1250) — compile-verified
//
#include <hip/hip_runtime.h>

typedef float v2f __attribute__((ext_vector_type(2)));
typedef float v8f __attribute__((ext_vector_type(8)));
typedef int   v4i __attribute__((ext_vector_type(4)));

#define BB 4
#define QL 512
#define KL 1024
#define QD 256
#define KD 256
#define VD 256
#define HD 128

// ---- CDNA5 async global->LDS staging (ASYNCcnt-tracked), with safe fallback
#if __has_builtin(__builtin_amdgcn_global_load_async_to_lds_b128)
#define HAVE_ASYNC_LDS 1
#define ASYNC_COPY16(gsrc, ldst)                                               \
    __builtin_amdgcn_global_load_async_to_lds_b128(                            \
        (__attribute__((address_space(1))) v4i*)(gsrc),                        \
        (__attribute__((address_space(3))) v4i*)(ldst), 0, 0)
#else
#define HAVE_ASYNC_LDS 0
#endif

#if __has_builtin(__builtin_amdgcn_s_wait_asynccnt)
#define ASYNC_WAIT() __builtin_amdgcn_s_wait_asynccnt(0)
#else
#define ASYNC_WAIT() asm volatile("s_wait_asynccnt 0x0" ::: "memory")
#endif

// ---------------------------------------------------------------------------
// Generic batched row-major f32 GEMM using V_WMMA_F32_16X16X4_F32 (wave32).
// Block = 128 threads = 4 waves; each wave computes one 16(M) x 16(N) C tile,
// block covers 16(M) x 64(N). Grid: (N/64, M/16, batch).
//
// wave32 VGPR layouts (CDNA5 ISA 7.12.2):
//   A 16x4 f32 : lanes 0-15 M=0..15 {K=0,K=1}, lanes 16-31 M=0..15 {K=2,K=3}
//   B 4x16 f32 : lanes 0-15 N=0..15 {K=0,K=1}, lanes 16-31 N=0..15 {K=2,K=3}
//   C 16x16 f32: VGPR r -> row (r + 8*half), col = lane&15
// ---------------------------------------------------------------------------
__global__ void gemm_f32_wmma(const float* __restrict__ A, const float* __restrict__ Bm,
                              float* __restrict__ C, int M, int N, int K,
                              int lda, int ldb, int ldc,
                              long sA, long sB, long sC) {
    const int wave = threadIdx.x >> 5;
    const int lane = threadIdx.x & 31;
    const int half = lane >> 4;     // 0: K=0,1   1: K=2,3
    const int l16  = lane & 15;
    const int m0 = blockIdx.y * 16;
    const int n0 = blockIdx.x * 64 + wave * 16;
    const int b  = blockIdx.z;
    if (m0 >= M || n0 >= N) return;   // wave-uniform: EXEC stays all-ones

    const float* Ab = A  + (long)b * sA;
    const float* Bb = Bm + (long)b * sB;
    float*       Cb = C  + (long)b * sC;

    v8f acc = {0.f, 0.f, 0.f, 0.f, 0.f, 0.f, 0.f, 0.f};

    const float* ap = Ab + (long)(m0 + l16) * lda + 2 * half;      // A[m][k + 2*half]
    const float* bp = Bb + (long)(2 * half) * ldb + n0 + l16;      // B[k + 2*half][n]

    #pragma unroll 4
    for (int k = 0; k < K; k += 4) {
        v2f a, bv;
        a.x  = ap[0];          // K = 2*half
        a.y  = ap[1];          // K = 2*half + 1
        bv.x = bp[0];
        bv.y = bp[ldb];
        acc = __builtin_amdgcn_wmma_f32_16x16x4_f32(
            /*neg_a=*/false, a, /*neg_b=*/false, bv,
            /*c_mod=*/(short)0, acc, /*reuse_a=*/false, /*reuse_b=*/false);
        ap += 4;
        bp += 4 * ldb;
    }

    #pragma unroll
    for (int r = 0; r < 8; ++r)
        Cb[(long)(m0 + r + 8 * half) * ldc + n0 + l16] = acc[r];
}

// ---------------------------------------------------------------------------
// energy[b,q,k] = sum_h we[h] * tanh(qp[b,q,h] + kp[b,k,h]),  masked.
//
// 32x32 (q,k) tile per 256-thread block, 2x2 outputs per thread (register
// blocking halves LDS bytes/element vs 1 output/thread). qp/kp tiles staged
// in LDS via GLOBAL_LOAD_ASYNC_TO_LDS_B128 (per-lane 16B chunks, ASYNCcnt),
// row stride padded to 132 floats: 16B-aligned rows, 4-bank skew so the 16
// distinct tk readers hit 16 distinct banks.
// tanh via HW transcendentals: tanh(x) = 1 - 2*rcp(exp2(2*log2e*x) + 1).
// ---------------------------------------------------------------------------
#define TQ 32
#define TK 32
#define LSTR 132

__global__ void energy_kernel(const float* __restrict__ qp, const float* __restrict__ kp,
                              const float* __restrict__ we, const int* __restrict__ mask,
                              float* __restrict__ energy) {
    __shared__ float qs[TQ * LSTR];
    __shared__ float ks[TK * LSTR];
    __shared__ float ws[HD];

    const int b  = blockIdx.z;
    const int q0 = blockIdx.y * TQ;
    const int k0 = blockIdx.x * TK;
    const int tid = threadIdx.x;            // 256 threads

#if HAVE_ASYNC_LDS
    // 32 rows x 32 chunks(16B) per tile = 1024 chunks; 4 per thread per tile.
    #pragma unroll
    for (int it = 0; it < 4; ++it) {
        const int c    = tid + it * 256;
        const int row  = c >> 5;            // 0..31
        const int col4 = (c & 31) * 4;      // 0,4,...,124
        ASYNC_COPY16(qp + ((long)b * QL + q0 + row) * HD + col4, &qs[row * LSTR + col4]);
        ASYNC_COPY16(kp + ((long)b * KL + k0 + row) * HD + col4, &ks[row * LSTR + col4]);
    }
#else
    for (int i = tid; i < TQ * HD; i += 256) {
        const int r = i >> 7, c = i & 127;
        qs[r * LSTR + c] = qp[((long)b * QL + q0 + r) * HD + c];
        ks[r * LSTR + c] = kp[((long)b * KL + k0 + r) * HD + c];
    }
#endif
    if (tid < HD) ws[tid] = we[tid];
    ASYNC_WAIT();
    __syncthreads();

    const int tq = tid >> 4;                // 0..15 -> rows tq, tq+16
    const int tk = tid & 15;                // 0..15 -> cols tk, tk+16
    const float* q0p = &qs[tq * LSTR];
    const float* q1p = &qs[(tq + 16) * LSTR];
    const float* k0p = &ks[tk * LSTR];
    const float* k1p = &ks[(tk + 16) * LSTR];

    float e00 = 0.f, e01 = 0.f, e10 = 0.f, e11 = 0.f;
    #pragma unroll 4
    for (int h = 0; h < HD; ++h) {
        const float w  = ws[h];
        const float a0 = q0p[h], a1 = q1p[h];
        const float c0 = k0p[h], c1 = k1p[h];
        const float C2 = 2.885390081777927f;          // 2/ln2
        const float t00 = 1.f - 2.f * __builtin_amdgcn_rcpf(__builtin_amdgcn_exp2f((a0 + c0) * C2) + 1.f);
        const float t01 = 1.f - 2.f * __builtin_amdgcn_rcpf(__builtin_amdgcn_exp2f((a0 + c1) * C2) + 1.f);
        const float t10 = 1.f - 2.f * __builtin_amdgcn_rcpf(__builtin_amdgcn_exp2f((a1 + c0) * C2) + 1.f);
        const float t11 = 1.f - 2.f * __builtin_amdgcn_rcpf(__builtin_amdgcn_exp2f((a1 + c1) * C2) + 1.f);
        e00 = fmaf(w, t00, e00);
        e01 = fmaf(w, t01, e01);
        e10 = fmaf(w, t10, e10);
        e11 = fmaf(w, t11, e11);
    }

    const long row0 = (long)b * QL + (q0 + tq);
    const long row1 = row0 + 16;
    const long i00 = row0 * KL + (k0 + tk);
    const long i01 = i00 + 16;
    const long i10 = row1 * KL + (k0 + tk);
    const long i11 = i10 + 16;
    const float NINF = -__builtin_inff();
    energy[i00] = (mask[i00] == 0) ? NINF : e00;
    energy[i01] = (mask[i01] == 0) ? NINF : e01;
    energy[i10] = (mask[i10] == 0) ? NINF : e10;
    energy[i11] = (mask[i11] == 0) ? NINF : e11;
}

// ---------------------------------------------------------------------------
// In-place row softmax over K=1024; one 128-thread block per (b,q) row.
// ---------------------------------------------------------------------------
__global__ void softmax_kernel(float* __restrict__ attn) {
    __shared__ float red[128];
    float* p = attn + (long)blockIdx.x * KL;
    const int tid = threadIdx.x;

    float v[8];
    float m = -__builtin_inff();
    #pragma unroll
    for (int i = 0; i < 8; ++i) { v[i] = p[tid + i * 128]; m = fmaxf(m, v[i]); }
    red[tid] = m; __syncthreads();
    for (int s = 64; s > 0; s >>= 1) {
        if (tid < s) red[tid] = fmaxf(red[tid], red[tid + s]);
        __syncthreads();
    }
    m = red[0]; __syncthreads();

    float sum = 0.f;
    #pragma unroll
    for (int i = 0; i < 8; ++i) {
        v[i] = __builtin_amdgcn_exp2f((v[i] - m) * 1.4426950408889634f);
        sum += v[i];
    }
    red[tid] = sum; __syncthreads();
    for (int s = 64; s > 0; s >>= 1) {
        if (tid < s) red[tid] += red[tid + s];
        __syncthreads();
    }
    const float inv = 1.0f / red[0];

    #pragma unroll
    for (int i = 0; i < 8; ++i) p[tid + i * 128] = v[i] * inv;
}

// ---------------------------------------------------------------------------
extern "C" void kernel_launch(void* const* d_in, const int* in_sizes, int n_in,
                              void* d_out, int out_size, void* d_ws, size_t ws_size,
                              hipStream_t stream) {
    const float* query = (const float*)d_in[0];   // (B,QL,QD)
    const float* key   = (const float*)d_in[1];   // (B,KL,KD)
    const float* value = (const float*)d_in[2];   // (B,KL,VD)
    const int*   mask  = (const int*)  d_in[3];   // (B,QL,KL)
    const float* Wq    = (const float*)d_in[4];   // (QD,HD)
    const float* Wk    = (const float*)d_in[5];   // (KD,HD)
    const float* we    = (const float*)d_in[6];   // (HD)

    float* context = (float*)d_out;                         // (B,QL,VD)
    float* attn    = context + (long)BB * QL * VD;          // (B,QL,KL)

    float* qp = (float*)d_ws;                               // (B*QL, HD) 1 MB
    float* kp = qp + (long)BB * QL * HD;                    // (B*KL, HD) 2 MB

    const dim3 blk(128);

    // qp = query @ Wq : M = B*QL = 2048, K = 256, N = 128
    gemm_f32_wmma<<<dim3(HD / 64, (BB * QL) / 16, 1), blk, 0, stream>>>(
        query, Wq, qp, BB * QL, HD, QD, QD, HD, HD, 0, 0, 0);

    // kp = key @ Wk : M = B*KL = 4096, K = 256, N = 128
    gemm_f32_wmma<<<dim3(HD / 64, (BB * KL) / 16, 1), blk, 0, stream>>>(
        key, Wk, kp, BB * KL, HD, KD, KD, HD, HD, 0, 0, 0);

    // masked energy -> attn buffer (in d_out)
    energy_kernel<<<dim3(KL / TK, QL / TQ, BB), dim3(256), 0, stream>>>(
        qp, kp, we, mask, attn);

    // softmax in place over K
    softmax_kernel<<<dim3(BB * QL), dim3(128), 0, stream>>>(attn);

    // context = attn @ value : batched, M = 512, N = 256, K = 1024
    gemm_f32_wmma<<<dim3(VD / 64, QL / 16, BB), blk, 0, stream>>>(
        attn, value, context, QL, VD, KL, KL, VD, VD,
        (long)QL * KL, (long)KL * VD, (long)QL * VD);
}